// MGPR_21680994910516
// MI455X (gfx1250) — compile-verified
//
#include <hip/hip_runtime.h>

// ---------------- problem dims (from reference) ----------------
#define NN   1000      // train points
#define DDIM 10        // output dims
#define PP   16        // input dims
#define NP   1024      // padded N = 64*16 (even # of 16-tiles -> exact 32x32 blocking)
#define NT   64        // 16x16 tiles per dim
#define NT2  32        // 32x32 tiles per dim
#define PAIRS (DDIM*DDIM)

typedef __attribute__((ext_vector_type(2))) float v2f;
typedef __attribute__((ext_vector_type(8))) float v8f;

// ---------------- workspace layout (floats) ----------------
static constexpr size_t SZ_BIG   = (size_t)DDIM * NP * NP;
static constexpr size_t OFF_B1   = 0;                               // Ky -> L -> iK
static constexpr size_t OFF_B2   = OFF_B1 + SZ_BIG;                 // Z = L^{-1}
static constexpr size_t OFF_NU   = OFF_B2 + SZ_BIG;                 // NP*PP
static constexpr size_t OFF_LI   = OFF_NU + (size_t)NP * PP;        // DDIM*PP
static constexpr size_t OFF_LS2  = OFF_LI + DDIM * PP;
static constexpr size_t OFF_SV   = OFF_LS2 + DDIM * PP;             // 16
static constexpr size_t OFF_NV   = OFF_SV + 16;                     // 16
static constexpr size_t OFF_SLI  = OFF_NV + 16;                     // DDIM*PP*PP
static constexpr size_t OFF_DETF = OFF_SLI + DDIM * PP * PP;        // 16
static constexpr size_t OFF_RS   = OFF_DETF + 16;                   // PAIRS*PP*PP
static constexpr size_t OFF_DETR = OFF_RS + (size_t)PAIRS * PP * PP;// 112
static constexpr size_t OFF_NUL  = OFF_DETR + 112;                  // DDIM*NP*PP
static constexpr size_t OFF_KM   = OFF_NUL + (size_t)DDIM * NP * PP;// DDIM*NP
static constexpr size_t OFF_Q    = OFF_KM + (size_t)DDIM * NP;
static constexpr size_t OFF_BETA = OFF_Q + (size_t)DDIM * NP;
static constexpr size_t OFF_TA   = OFF_BETA + (size_t)DDIM * NP;    // PAIRS*NP*PP
static constexpr size_t OFF_TB   = OFF_TA + (size_t)PAIRS * NP * PP;
static constexpr size_t OFF_QAA  = OFF_TB + (size_t)PAIRS * NP * PP;// PAIRS*NP
static constexpr size_t OFF_QBB  = OFF_QAA + (size_t)PAIRS * NP;
static constexpr size_t OFF_M    = OFF_QBB + (size_t)PAIRS * NP;    // 16
static constexpr size_t OFF_SACC = OFF_M + 16;                      // 112
static constexpr size_t OFF_TR   = OFF_SACC + 112;                  // 16
static constexpr size_t OFF_W    = OFF_TR + 16;                     // DDIM*PP

// ---------------- tiny dense linear algebra (per-thread) ----------------
__device__ float det16_lu(float A[PP][PP]) {
    float det = 1.f;
    for (int k = 0; k < PP; ++k) {
        float piv = A[k][k];
        det *= piv;
        float ip = 1.f / piv;
        for (int i = k + 1; i < PP; ++i) {
            float f = A[i][k] * ip;
            for (int j = k + 1; j < PP; ++j) A[i][j] -= f * A[k][j];
        }
    }
    return det;
}

__device__ void inv16_gj(float A[PP][PP], float Inv[PP][PP]) {
    for (int i = 0; i < PP; ++i)
        for (int j = 0; j < PP; ++j) Inv[i][j] = (i == j) ? 1.f : 0.f;
    for (int k = 0; k < PP; ++k) {
        float ip = 1.f / A[k][k];
        for (int j = 0; j < PP; ++j) { A[k][j] *= ip; Inv[k][j] *= ip; }
        for (int i = 0; i < PP; ++i) {
            if (i == k) continue;
            float f = A[i][k];
            for (int j = 0; j < PP; ++j) { A[i][j] -= f * A[k][j]; Inv[i][j] -= f * Inv[k][j]; }
        }
    }
}

__device__ float wave_sum(float v) {
    for (int off = 16; off > 0; off >>= 1) v += __shfl_down(v, off, 32);
    return v;
}

// ---------------- kernels ----------------
__global__ void k_init(float* Sacc, float* tracc) {
    int idx = blockIdx.x * blockDim.x + threadIdx.x;
    if (idx < 112) Sacc[idx] = 0.f;
    if (idx < 16)  tracc[idx] = 0.f;
}

__global__ void k_prep(const float* X, const float* log_ls, const float* lsv,
                       const float* lnv, const float* m_in,
                       float* nu, float* Li, float* ls2, float* sv, float* nv) {
    int idx = blockIdx.x * blockDim.x + threadIdx.x;
    if (idx < NP * PP) {
        int n = idx / PP, p = idx % PP;
        nu[idx] = (n < NN) ? (X[n * PP + p] - m_in[p]) : 0.f;
    }
    if (idx < DDIM * PP) {
        float l = log_ls[idx];
        Li[idx]  = __expf(-2.f * l);   // 1/ls^2
        ls2[idx] = __expf( 2.f * l);   // ls^2
    }
    if (idx < DDIM) {
        sv[idx] = __expf(lsv[idx]);
        nv[idx] = __expf(lnv[idx]);
    }
}

// Ky[a] = sv*exp(-0.5 d2) + (nv+1e-6) I  ; padded rows/cols -> identity
__global__ void k_ky(const float* X, const float* Li, const float* sv,
                     const float* nv, float* B1) {
    int i = blockIdx.x * 16 + threadIdx.x;
    int j = blockIdx.y * 16 + threadIdx.y;
    int a = blockIdx.z;
    float val;
    if (i < NN && j < NN) {
        float d2 = 0.f;
        #pragma unroll
        for (int p = 0; p < PP; ++p) {
            float d = X[i * PP + p] - X[j * PP + p];
            d2 += Li[a * PP + p] * d * d;
        }
        val = sv[a] * __expf(-0.5f * d2);
        if (i == j) val += nv[a] + 1e-6f;
    } else {
        val = (i == j) ? 1.f : 0.f;
    }
    B1[(size_t)a * NP * NP + (size_t)i * NP + j] = val;
}

// in-place Cholesky (lower), one workgroup per a, panel column cached in LDS
__global__ void __launch_bounds__(1024) k_chol(float* B1) {
    float* A = B1 + (size_t)blockIdx.x * NP * NP;
    __shared__ float col[NP];
    int t = threadIdx.x, nt = blockDim.x;
    for (int k = 0; k < NP; ++k) {
        if (t == 0) A[(size_t)k * NP + k] = sqrtf(A[(size_t)k * NP + k]);
        __syncthreads();
        float d = A[(size_t)k * NP + k];
        for (int i = k + 1 + t; i < NP; i += nt) {
            float v = A[(size_t)i * NP + k] / d;
            A[(size_t)i * NP + k] = v;
            col[i] = v;
        }
        __syncthreads();
        for (int j = k + 1; j < NP; ++j) {
            float ljk = col[j];
            for (int i = j + t; i < NP; i += nt)
                A[(size_t)i * NP + j] -= col[i] * ljk;
        }
        __syncthreads();
    }
}

// Z = L^{-1} (lower triangular inverse), one WAVE per column:
// lanes parallelize the forward-substitution dot, lane0 writes.
// (same-wave store->load same-address ordering is guaranteed by HW)
__global__ void __launch_bounds__(1024) k_zinv(const float* B1, float* B2) {
    const float* L = B1 + (size_t)blockIdx.x * NP * NP;
    float* Z = B2 + (size_t)blockIdx.x * NP * NP;
    int wave = threadIdx.x >> 5;
    int lane = threadIdx.x & 31;
    int nwaves = blockDim.x >> 5;
    for (int j = wave; j < NP; j += nwaves) {
        for (int i = lane; i < j; i += 32) Z[(size_t)i * NP + j] = 0.f;
        if (lane == 0) Z[(size_t)j * NP + j] = 1.f / L[(size_t)j * NP + j];
        for (int i = j + 1; i < NP; ++i) {
            float s = 0.f;
            for (int k = j + lane; k < i; k += 32)
                s += L[(size_t)i * NP + k] * Z[(size_t)k * NP + j];
            s = wave_sum(s);
            if (lane == 0) Z[(size_t)i * NP + j] = -s / L[(size_t)i * NP + i];
        }
    }
}

// iK = Z^T Z via f32 WMMA, one wave per 32x32 output block (2x2 tile blocking)
__global__ void __launch_bounds__(32) k_syrk(const float* B2, float* B1) {
    int a = blockIdx.y;
    int ti = blockIdx.x / NT2, tj = blockIdx.x % NT2;
    int n0 = ti * 32, m0 = tj * 32;
    const float* Z = B2 + (size_t)a * NP * NP;
    float* iK = B1 + (size_t)a * NP * NP;
    int lane = threadIdx.x;
    int c = lane & 15;            // M (A) / N (B) index within 16-tile
    int kh = (lane >> 4) * 2;     // which K-pair this half-wave holds
    v8f acc00 = {}, acc01 = {}, acc10 = {}, acc11 = {};
    int k0 = 32 * (ti > tj ? ti : tj);   // Z[k,n]==0 for k<n
    for (int k = k0; k < NP; k += 4) {
        const float* r0 = Z + (size_t)(k + kh) * NP;
        const float* r1 = Z + (size_t)(k + kh + 1) * NP;
        v2f a0, a1, b0, b1;
        a0.x = r0[n0 + c];      a0.y = r1[n0 + c];
        a1.x = r0[n0 + 16 + c]; a1.y = r1[n0 + 16 + c];
        b0.x = r0[m0 + c];      b0.y = r1[m0 + c];
        b1.x = r0[m0 + 16 + c]; b1.y = r1[m0 + 16 + c];
        acc00 = __builtin_amdgcn_wmma_f32_16x16x4_f32(false, a0, false, b0, (short)0, acc00, false, false);
        acc01 = __builtin_amdgcn_wmma_f32_16x16x4_f32(false, a0, false, b1, (short)0, acc01, false, false);
        acc10 = __builtin_amdgcn_wmma_f32_16x16x4_f32(false, a1, false, b0, (short)0, acc10, false, false);
        acc11 = __builtin_amdgcn_wmma_f32_16x16x4_f32(false, a1, false, b1, (short)0, acc11, false, false);
    }
    int rbase = (lane >> 4) * 8;
    #pragma unroll
    for (int r = 0; r < 8; ++r) {
        int row0 = n0 + rbase + r;
        int row1 = row0 + 16;
        iK[(size_t)row0 * NP + m0 + c]      = acc00[r];
        iK[(size_t)row0 * NP + m0 + 16 + c] = acc01[r];
        iK[(size_t)row1 * NP + m0 + c]      = acc10[r];
        iK[(size_t)row1 * NP + m0 + 16 + c] = acc11[r];
    }
}

// beta[a,n] = sum_m iK[a,n,m] * Y[m,a]   (one wave per row, coalesced)
__global__ void k_beta(const float* B1, const float* Y, float* beta) {
    int gw = (int)((blockIdx.x * (size_t)blockDim.x + threadIdx.x) >> 5);
    int lane = threadIdx.x & 31;
    if (gw >= DDIM * NP) return;
    int a = gw / NP, n = gw % NP;
    float s = 0.f;
    if (n < NN) {
        const float* row = B1 + (size_t)a * NP * NP + (size_t)n * NP;
        for (int m = lane; m < NN; m += 32) s += row[m] * Y[m * DDIM + a];
    }
    s = wave_sum(s);
    if (lane == 0) beta[gw] = s;
}

// sL = s + diag(ls^2): det_factor + inverse (one thread per a)
__global__ void k_sl(const float* s_in, const float* ls2, float* sLinv, float* detf) {
    int a = threadIdx.x;
    if (a >= DDIM) return;
    float A[PP][PP], B[PP][PP], Inv[PP][PP];
    float prod = 1.f;
    for (int i = 0; i < PP; ++i) {
        prod *= ls2[a * PP + i];
        for (int j = 0; j < PP; ++j)
            A[i][j] = s_in[i * PP + j] + ((i == j) ? ls2[a * PP + i] : 0.f);
    }
    for (int i = 0; i < PP; ++i)
        for (int j = 0; j < PP; ++j) B[i][j] = A[i][j];
    float det = det16_lu(B);
    detf[a] = det / prod;
    for (int i = 0; i < PP; ++i) A[i][i] += 1e-8f;
    inv16_gj(A, Inv);
    for (int i = 0; i < PP; ++i)
        for (int j = 0; j < PP; ++j)
            sLinv[(a * PP + i) * PP + j] = Inv[i][j];
}

// R[a,b], det_R, Rs = R^{-1} s  (one thread per pair)
__global__ void k_R(const float* s_in, const float* Li, float* Rs, float* detR) {
    int pr = blockIdx.x * blockDim.x + threadIdx.x;
    if (pr >= PAIRS) return;
    int a = pr / DDIM, b = pr % DDIM;
    float R[PP][PP], B[PP][PP], Inv[PP][PP];
    for (int i = 0; i < PP; ++i)
        for (int j = 0; j < PP; ++j)
            R[i][j] = s_in[i * PP + j] * (Li[a * PP + j] + Li[b * PP + j]) +
                      ((i == j) ? 1.f : 0.f);
    for (int i = 0; i < PP; ++i)
        for (int j = 0; j < PP; ++j) B[i][j] = R[i][j];
    detR[pr] = det16_lu(B);
    for (int i = 0; i < PP; ++i) R[i][i] += 1e-8f;
    inv16_gj(R, Inv);
    for (int i = 0; i < PP; ++i)
        for (int j = 0; j < PP; ++j) {
            float acc = 0.f;
            for (int k = 0; k < PP; ++k) acc += Inv[i][k] * s_in[k * PP + j];
            Rs[((size_t)pr * PP + i) * PP + j] = acc;
        }
}

// nuL[a,n,:] = nu[n,:]*Li[a,:] ; k_m[a,n] (0 at pad rows)
__global__ void k_nul(const float* nu, const float* Li, const float* sv,
                      float* nuL, float* km) {
    int idx = blockIdx.x * blockDim.x + threadIdx.x;
    if (idx >= DDIM * NP) return;
    int a = idx / NP, n = idx % NP;
    float dot = 0.f;
    #pragma unroll
    for (int p = 0; p < PP; ++p) {
        float v = nu[n * PP + p] * Li[a * PP + p];
        nuL[((size_t)a * NP + n) * PP + p] = v;
        dot += nu[n * PP + p] * v;
    }
    km[idx] = (n < NN) ? sv[a] * __expf(-0.5f * dot) : 0.f;
}

// q[a,n] = sv/sqrt(max(detf,1e-12)) * exp(-0.5 * nu sLinv nu)
__global__ void k_qmean(const float* nu, const float* sLinv, const float* detf,
                        const float* sv, float* q) {
    int idx = blockIdx.x * blockDim.x + threadIdx.x;
    if (idx >= DDIM * NP) return;
    int a = idx / NP, n = idx % NP;
    float val = 0.f;
    if (n < NN) {
        float quad = 0.f;
        for (int p = 0; p < PP; ++p) {
            float inner = 0.f;
            #pragma unroll
            for (int r = 0; r < PP; ++r)
                inner += sLinv[(a * PP + p) * PP + r] * nu[n * PP + r];
            quad += nu[n * PP + p] * inner;
        }
        val = sv[a] * rsqrtf(fmaxf(detf[a], 1e-12f)) * __expf(-0.5f * quad);
    }
    q[idx] = val;
}

// t_a[pr,n,p] = nuL[a,n,:] . Rs[pr,:,p] ; t_b with nuL[b]
__global__ void k_t(const float* nuL, const float* Rs, float* ta, float* tb) {
    size_t idx = (size_t)blockIdx.x * blockDim.x + threadIdx.x;
    if (idx >= (size_t)PAIRS * NP * PP) return;
    int p = (int)(idx % PP);
    int n = (int)((idx / PP) % NP);
    int pr = (int)(idx / ((size_t)PP * NP));
    int a = pr / DDIM, b = pr % DDIM;
    const float* R = Rs + (size_t)pr * PP * PP;
    float sa = 0.f, sb = 0.f;
    #pragma unroll
    for (int qq = 0; qq < PP; ++qq) {
        float rq = R[qq * PP + p];
        sa += nuL[((size_t)a * NP + n) * PP + qq] * rq;
        sb += nuL[((size_t)b * NP + n) * PP + qq] * rq;
    }
    ta[idx] = sa;
    tb[idx] = sb;
}

// q_aa[pr,n], q_bb[pr,n]
__global__ void k_qab(const float* nuL, const float* ta, const float* tb,
                      float* qaa, float* qbb) {
    int idx = blockIdx.x * blockDim.x + threadIdx.x;
    if (idx >= PAIRS * NP) return;
    int n = idx % NP, pr = idx / NP;
    int a = pr / DDIM, b = pr % DDIM;
    float sa = 0.f, sb = 0.f;
    #pragma unroll
    for (int p = 0; p < PP; ++p) {
        sa += nuL[((size_t)a * NP + n) * PP + p] * ta[((size_t)pr * NP + n) * PP + p];
        sb += nuL[((size_t)b * NP + n) * PP + p] * tb[((size_t)pr * NP + n) * PP + p];
    }
    qaa[idx] = sa;
    qbb[idx] = sb;
}

// M[a] = sum beta*q ; w[a,p] = sum beta*q*nu
__global__ void k_mw(const float* beta, const float* q, const float* nu,
                     float* M, float* w) {
    int idx = blockIdx.x * blockDim.x + threadIdx.x;
    if (idx < DDIM) {
        float s = 0.f;
        for (int n = 0; n < NN; ++n)
            s += beta[(size_t)idx * NP + n] * q[(size_t)idx * NP + n];
        M[idx] = s;
    } else if (idx < DDIM + DDIM * PP) {
        int k = idx - DDIM;
        int a = k / PP, p = k % PP;
        float s = 0.f;
        for (int n = 0; n < NN; ++n)
            s += beta[(size_t)a * NP + n] * q[(size_t)a * NP + n] * nu[n * PP + p];
        w[k] = s;
    }
}

// Q tile via WMMA + fused beta^T Q beta and tr(iK Q) accumulation
__global__ void __launch_bounds__(32) k_Qs(const float* nuL, const float* tb,
                                           const float* km, const float* qaa,
                                           const float* qbb, const float* detR,
                                           const float* beta, const float* B1,
                                           float* Sacc, float* tracc) {
    int pr = blockIdx.y;
    int a = pr / DDIM, b = pr % DDIM;
    int ti = blockIdx.x / NT, tj = blockIdx.x % NT;
    int n0 = ti * 16, m0 = tj * 16;
    int lane = threadIdx.x;
    int c = lane & 15;
    int kh = (lane >> 4) * 2;
    const float* nuLa = nuL + (size_t)a * NP * PP;
    const float* tbp  = tb  + (size_t)pr * NP * PP;
    v8f acc = {};
    #pragma unroll
    for (int kk = 0; kk < PP; kk += 4) {
        v2f av, bv;
        av.x = nuLa[(size_t)(n0 + c) * PP + kk + kh];
        av.y = nuLa[(size_t)(n0 + c) * PP + kk + kh + 1];
        bv.x = tbp [(size_t)(m0 + c) * PP + kk + kh];
        bv.y = tbp [(size_t)(m0 + c) * PP + kk + kh + 1];
        acc = __builtin_amdgcn_wmma_f32_16x16x4_f32(
            false, av, false, bv, (short)0, acc, false, false);
    }
    float invsq = rsqrtf(fmaxf(detR[pr], 1e-12f));
    float kmb   = km  [(size_t)b * NP + m0 + c];
    float qbbv  = qbb [(size_t)pr * NP + m0 + c];
    float betab = beta[(size_t)b * NP + m0 + c];
    const float* kma   = km   + (size_t)a * NP;
    const float* qaap  = qaa  + (size_t)pr * NP;
    const float* betaa = beta + (size_t)a * NP;
    const float* iKa   = B1   + (size_t)a * NP * NP;
    float spart = 0.f, trpart = 0.f;
    int rbase = (lane >> 4) * 8;
    #pragma unroll
    for (int r = 0; r < 8; ++r) {
        int row = n0 + rbase + r;
        float Qv = kma[row] * kmb * invsq *
                   __expf(0.5f * (qaap[row] + qbbv + 2.f * acc[r]));
        spart += betaa[row] * Qv * betab;
        if (a == b) trpart += iKa[(size_t)row * NP + m0 + c] * Qv;
    }
    spart = wave_sum(spart);
    trpart = wave_sum(trpart);
    if (lane == 0) {
        atomicAdd(&Sacc[pr], spart);
        if (a == b) atomicAdd(&tracc[a], trpart);
    }
}

// out = [M(10) | S(100) | V(16x10)]
__global__ void k_final(const float* M, const float* Sacc, const float* tracc,
                        const float* sv, const float* s_in, const float* sLinv,
                        const float* w, float* out) {
    int idx = blockIdx.x * blockDim.x + threadIdx.x;
    if (idx < DDIM) {
        out[idx] = M[idx];
    } else if (idx < DDIM + PAIRS) {
        int k = idx - DDIM;
        int a = k / DDIM, b = k % DDIM;
        float v = Sacc[k] - M[a] * M[b];
        if (a == b) v += sv[a] - tracc[a];
        out[idx] = v;
    } else if (idx < DDIM + PAIRS + PP * DDIM) {
        int k = idx - DDIM - PAIRS;
        int p = k / DDIM, a = k % DDIM;
        float acc = 0.f;
        for (int qq = 0; qq < PP; ++qq) {
            float inner = 0.f;
            #pragma unroll
            for (int r = 0; r < PP; ++r)
                inner += sLinv[(a * PP + qq) * PP + r] * w[a * PP + r];
            acc += s_in[p * PP + qq] * inner;
        }
        out[idx] = acc;
    }
}

extern "C" void kernel_launch(void* const* d_in, const int* in_sizes, int n_in,
                              void* d_out, int out_size, void* d_ws, size_t ws_size,
                              hipStream_t stream) {
    const float* X   = (const float*)d_in[0];
    const float* Y   = (const float*)d_in[1];
    const float* lls = (const float*)d_in[2];
    const float* lsv = (const float*)d_in[3];
    const float* lnv = (const float*)d_in[4];
    const float* m_in = (const float*)d_in[5];
    const float* s_in = (const float*)d_in[6];
    float* out = (float*)d_out;
    float* ws = (float*)d_ws;

    float* B1   = ws + OFF_B1;
    float* B2   = ws + OFF_B2;
    float* nu   = ws + OFF_NU;
    float* Li   = ws + OFF_LI;
    float* ls2  = ws + OFF_LS2;
    float* sv   = ws + OFF_SV;
    float* nv   = ws + OFF_NV;
    float* sLinv= ws + OFF_SLI;
    float* detf = ws + OFF_DETF;
    float* Rs   = ws + OFF_RS;
    float* detR = ws + OFF_DETR;
    float* nuL  = ws + OFF_NUL;
    float* km   = ws + OFF_KM;
    float* q    = ws + OFF_Q;
    float* beta = ws + OFF_BETA;
    float* ta   = ws + OFF_TA;
    float* tbuf = ws + OFF_TB;
    float* qaa  = ws + OFF_QAA;
    float* qbb  = ws + OFF_QBB;
    float* M    = ws + OFF_M;
    float* Sacc = ws + OFF_SACC;
    float* trac = ws + OFF_TR;
    float* w    = ws + OFF_W;

    k_init<<<1, 128, 0, stream>>>(Sacc, trac);
    k_prep<<<(NP * PP + 255) / 256, 256, 0, stream>>>(X, lls, lsv, lnv, m_in,
                                                      nu, Li, ls2, sv, nv);
    k_ky<<<dim3(NT, NT, DDIM), dim3(16, 16), 0, stream>>>(X, Li, sv, nv, B1);
    k_chol<<<DDIM, 1024, 0, stream>>>(B1);
    k_zinv<<<DDIM, 1024, 0, stream>>>(B1, B2);
    k_syrk<<<dim3(NT2 * NT2, DDIM), 32, 0, stream>>>(B2, B1);
    k_beta<<<(DDIM * NP * 32 + 255) / 256, 256, 0, stream>>>(B1, Y, beta);
    k_sl<<<1, 32, 0, stream>>>(s_in, ls2, sLinv, detf);
    k_R<<<1, 128, 0, stream>>>(s_in, Li, Rs, detR);
    k_nul<<<(DDIM * NP + 255) / 256, 256, 0, stream>>>(nu, Li, sv, nuL, km);
    k_qmean<<<(DDIM * NP + 255) / 256, 256, 0, stream>>>(nu, sLinv, detf, sv, q);
    {
        size_t total = (size_t)PAIRS * NP * PP;
        k_t<<<(unsigned)((total + 255) / 256), 256, 0, stream>>>(nuL, Rs, ta, tbuf);
    }
    k_qab<<<(PAIRS * NP + 255) / 256, 256, 0, stream>>>(nuL, ta, tbuf, qaa, qbb);
    k_mw<<<1, 256, 0, stream>>>(beta, q, nu, M, w);
    k_Qs<<<dim3(NT * NT, PAIRS), 32, 0, stream>>>(nuL, tbuf, km, qaa, qbb,
                                                  detR, beta, B1, Sacc, trac);
    k_final<<<2, 256, 0, stream>>>(M, Sacc, trac, sv, s_in, sLinv, w, out);
}